// CLUB_NCE_89781996356068
// MI455X (gfx1250) — compile-verified
//
#include <hip/hip_runtime.h>
#include <math.h>

typedef __attribute__((ext_vector_type(2))) float v2f;
typedef __attribute__((ext_vector_type(8))) float v8f;

#define NS 512   // samples
#define DD 400   // input dim
#define HH 400   // hidden dim
#define W1STRIDE (2 * DD)

// ---------------------------------------------------------------------------
// Stable softplus
// ---------------------------------------------------------------------------
__device__ __forceinline__ float softplus_f(float z) {
    float az = fabsf(z);
    return fmaxf(z, 0.0f) + log1pf(__expf(-az));
}

// ---------------------------------------------------------------------------
// Kernel A: f32 WMMA GEMM.
//   hx  = x @ W1[:, :D]^T
//   hyb = y @ W1[:, D:]^T + b1   (bias folded in, broadcast over rows)
// One wave computes one 16x16 output tile via V_WMMA_F32_16X16X4_F32,
// looping K over D in steps of 4 (100 WMMAs per tile).
// 2 matrices * (512/16)*(400/16) = 1600 tiles; 8 waves/block -> 200 blocks.
// ---------------------------------------------------------------------------
__global__ void club_gemm_wmma(const float* __restrict__ x,
                               const float* __restrict__ y,
                               const float* __restrict__ W1,
                               const float* __restrict__ b1,
                               float* __restrict__ hx,
                               float* __restrict__ hyb) {
    const int wave = (blockIdx.x * blockDim.x + threadIdx.x) >> 5;
    const int lane = threadIdx.x & 31;

    const int mat  = wave / 800;           // 0: x -> hx, 1: y -> hyb
    const int tile = wave - mat * 800;
    const int tr   = tile / 25;            // row tile (0..31)
    const int tc   = tile - tr * 25;       // col tile (0..24)
    const int r0   = tr << 4;
    const int c0   = tc << 4;

    const float* A   = mat ? y : x;
    const int wcol0  = mat ? DD : 0;       // W1y starts at column D

    // A-matrix 16x4 f32 layout: lanes 0-15 rows M, VGPR0/1 = K {0,1};
    // lanes 16-31 same rows, K {2,3}.  B 4x16: lane = N, VGPR0/1 = K pair.
    const int mrow  = lane & 15;           // M for A, N for B/C/D
    const int khalf = (lane >> 4) << 1;    // 0 or 2

    const float* arow = A  + (size_t)(r0 + mrow) * DD + khalf;
    const float* brow = W1 + (size_t)(c0 + mrow) * W1STRIDE + wcol0 + khalf;

    v8f c = {};
    for (int k = 0; k < DD; k += 4) {
        v2f a, b;
        a.x = arow[k];     a.y = arow[k + 1];
        b.x = brow[k];     b.y = brow[k + 1];
        c = __builtin_amdgcn_wmma_f32_16x16x4_f32(
                /*neg_a=*/false, a, /*neg_b=*/false, b,
                /*c_mod=*/(short)0, c, /*reuse_a=*/false, /*reuse_b=*/false);
    }

    const float bias  = mat ? b1[c0 + mrow] : 0.0f;  // column bias (n = mrow)
    float*      Out   = mat ? hyb : hx;
    const int   mbase = (lane >> 4) << 3;            // 0 or 8

    // C/D layout: VGPR v -> M = v + mbase, N = lane&15
#pragma unroll
    for (int v = 0; v < 8; ++v) {
        Out[(size_t)(r0 + mbase + v) * HH + c0 + mrow] = c[v] + bias;
    }
}

// ---------------------------------------------------------------------------
// Kernel B: pairwise T1 with fused online logsumexp + sum.
// Block owns TI=8 i-rows (hy' tile in LDS, broadcast reads per wave) and
// streams hx in TJ=32-row LDS chunks (stride padded to 401: conflict-free).
// Thread (wave ii, lane jj) owns pair (i0+ii, ch*32+jj) per chunk.
// ---------------------------------------------------------------------------
#define TI 8
#define TJ 32
#define HXP (HH + 1)

__global__ void club_pairwise(const float* __restrict__ hx,
                              const float* __restrict__ hyb,
                              const float* __restrict__ W2,
                              const float* __restrict__ b2p,
                              float* __restrict__ lse,
                              float* __restrict__ rowsum) {
    __shared__ float sHy[TI * HH];    // 12800 B
    __shared__ float sHx[TJ * HXP];   // 51328 B   (total 64128 B)

    const int tid = threadIdx.x;
    const int ii  = tid >> 5;         // wave id == local i index
    const int jj  = tid & 31;
    const int i0  = blockIdx.x * TI;
    const float b2 = b2p[0];

    // hy' rows i0..i0+7 are contiguous: flat coalesced copy
    for (int idx = tid; idx < TI * HH; idx += blockDim.x)
        sHy[idx] = hyb[(size_t)i0 * HH + idx];

    float m = -INFINITY, s = 0.0f, tsum = 0.0f;

    for (int ch = 0; ch < NS / TJ; ++ch) {
        __syncthreads();
        const int j0 = ch * TJ;
        for (int idx = tid; idx < TJ * HH; idx += blockDim.x) {
            const int r  = idx / HH;
            const int cc = idx - r * HH;
            sHx[r * HXP + cc] = hx[(size_t)(j0 + r) * HH + cc];
        }
        __syncthreads();

        // prefetch next chunk into cache (global_prefetch_b8)
        if (ch + 1 < NS / TJ)
            __builtin_prefetch(&hx[(size_t)(j0 + TJ) * HH + tid * 50], 0, 3);

        const float* hyrow = &sHy[ii * HH];    // broadcast within wave
        const float* hxrow = &sHx[jj * HXP];   // padded: no bank conflicts

        float acc = 0.0f;
#pragma unroll 4
        for (int h = 0; h < HH; ++h) {
            const float v = hyrow[h] + hxrow[h];      // b1 already folded in
            acc = fmaf(W2[h], fmaxf(v, 0.0f), acc);   // W2[h] wave-uniform
        }
        const float t1 = softplus_f(acc + b2);
        tsum += t1;
        if (t1 > m) { s = s * __expf(m - t1) + 1.0f; m = t1; }
        else        { s += __expf(t1 - m); }
    }

    // merge (m,s) and tsum across the 32 lanes of this wave (all same i)
#pragma unroll
    for (int off = 16; off > 0; off >>= 1) {
        const float m2 = __shfl_xor(m, off, 32);
        const float s2 = __shfl_xor(s, off, 32);
        const float M  = fmaxf(m, m2);
        s = s * __expf(m - M) + s2 * __expf(m2 - M);
        m = M;
        tsum += __shfl_xor(tsum, off, 32);
    }
    if (jj == 0) {
        lse[i0 + ii]    = m + logf(s);
        rowsum[i0 + ii] = tsum;
    }
}

// ---------------------------------------------------------------------------
// Kernel C: diagonal scores T0[i] = softplus(sum_h W2[h]*relu(hx[i]+hy'[i])+b2)
// One wave per row, lanes stride over h.
// ---------------------------------------------------------------------------
__global__ void club_t0(const float* __restrict__ hx,
                        const float* __restrict__ hyb,
                        const float* __restrict__ W2,
                        const float* __restrict__ b2p,
                        float* __restrict__ T0) {
    const int wave = (blockIdx.x * blockDim.x + threadIdx.x) >> 5;
    const int lane = threadIdx.x & 31;
    if (wave >= NS) return;

    float acc = 0.0f;
    for (int h = lane; h < HH; h += 32) {
        const float v = hx[(size_t)wave * HH + h] + hyb[(size_t)wave * HH + h];
        acc = fmaf(W2[h], fmaxf(v, 0.0f), acc);
    }
#pragma unroll
    for (int off = 16; off > 0; off >>= 1)
        acc += __shfl_xor(acc, off, 32);
    if (lane == 0) T0[wave] = softplus_f(acc + b2p[0]);
}

// ---------------------------------------------------------------------------
// Kernel D: final reduction -> (lower_bound, upper_bound)
// Single block, 512 threads (16 waves).
// ---------------------------------------------------------------------------
__global__ void club_finalize(const float* __restrict__ T0,
                              const float* __restrict__ lse,
                              const float* __restrict__ rowsum,
                              float* __restrict__ out) {
    __shared__ float sA[16], sB[16], sC[16];
    const int t = threadIdx.x;
    float a = T0[t], b = lse[t], c = rowsum[t];
#pragma unroll
    for (int off = 16; off > 0; off >>= 1) {
        a += __shfl_xor(a, off, 32);
        b += __shfl_xor(b, off, 32);
        c += __shfl_xor(c, off, 32);
    }
    const int w = t >> 5, lane = t & 31;
    if (lane == 0) { sA[w] = a; sB[w] = b; sC[w] = c; }
    __syncthreads();
    if (t == 0) {
        float ta = 0.0f, tb = 0.0f, tc = 0.0f;
        for (int k = 0; k < 16; ++k) { ta += sA[k]; tb += sB[k]; tc += sC[k]; }
        const float t0_mean = ta / (float)NS;
        out[0] = t0_mean - (tb / (float)NS - logf((float)NS));   // lower
        out[1] = t0_mean - tc / ((float)NS * (float)NS);         // upper
    }
}

// ---------------------------------------------------------------------------
// Host launch
// ---------------------------------------------------------------------------
extern "C" void kernel_launch(void* const* d_in, const int* in_sizes, int n_in,
                              void* d_out, int out_size, void* d_ws, size_t ws_size,
                              hipStream_t stream) {
    (void)in_sizes; (void)n_in; (void)out_size; (void)ws_size;

    const float* x  = (const float*)d_in[0];   // (512, 400)
    const float* y  = (const float*)d_in[1];   // (512, 400)
    const float* W1 = (const float*)d_in[2];   // (400, 800)
    const float* b1 = (const float*)d_in[3];   // (400,)
    const float* W2 = (const float*)d_in[4];   // (1, 400)
    const float* b2 = (const float*)d_in[5];   // (1,)
    float* out = (float*)d_out;                // 2 floats

    float* ws     = (float*)d_ws;
    float* hx     = ws;                        // 512*400
    float* hyb    = ws + (size_t)NS * HH;      // 512*400  (hy + b1)
    float* lse    = hyb + (size_t)NS * HH;     // 512
    float* rowsum = lse + NS;                  // 512
    float* T0     = rowsum + NS;               // 512

    // A: 1600 wave-tiles, 8 waves/block
    club_gemm_wmma<<<200, 256, 0, stream>>>(x, y, W1, b1, hx, hyb);
    // B: 64 blocks * 8 i-rows = 512 rows
    club_pairwise<<<NS / TI, 256, 0, stream>>>(hx, hyb, W2, b2, lse, rowsum);
    // C: 512 waves
    club_t0<<<64, 256, 0, stream>>>(hx, hyb, W2, b2, T0);
    // D: final scalars
    club_finalize<<<1, 512, 0, stream>>>(T0, lse, rowsum, out);
}